// Controller_4226247819587
// MI455X (gfx1250) — compile-verified
//
#include <hip/hip_runtime.h>

typedef float v2f __attribute__((ext_vector_type(2)));
typedef float v8f __attribute__((ext_vector_type(8)));

#define DFEAT 128
#define BN_EPS 1e-5f

// ---------------- degree / normalization ----------------
__global__ void deg_init_kernel(float* __restrict__ deg, int N) {
    int i = blockIdx.x * blockDim.x + threadIdx.x;
    if (i < N) deg[i] = 1.0f;                 // self-loop contributes 1
}

__global__ void deg_accum_kernel(const long long* __restrict__ dst, float* __restrict__ deg, int E) {
    int e = blockIdx.x * blockDim.x + threadIdx.x;
    if (e < E) unsafeAtomicAdd(&deg[(int)dst[e]], 1.0f);
}

__global__ void dinv_kernel(float* __restrict__ deg, int N) {
    int i = blockIdx.x * blockDim.x + threadIdx.x;
    if (i < N) deg[i] = rsqrtf(fmaxf(deg[i], 1.0f));   // deg buffer becomes dinv in place
}

// ---------------- WMMA f32 GEMM: H[N,128] = X[N,128] @ W[128,128] ----------------
// 256 threads = 8 waves per block. Block b covers rows [16b,16b+16); wave w covers cols [16w,16w+16).
// Per wave: 32 x V_WMMA_F32_16X16X4_F32, K advancing by 4 each step.
// Grid is exactly N/16 blocks (N is a multiple of 16) -> no bounds checks anywhere.
__global__ void gemm_wmma_kernel(const float* __restrict__ X, const float* __restrict__ W,
                                 float* __restrict__ H) {
    const int wave = threadIdx.x >> 5;
    const int lane = threadIdx.x & 31;
    const int row0 = blockIdx.x * 16;
    const int col0 = wave * 16;

    const int m   = lane & 15;            // A-matrix row within tile (both lane halves)
    const int n   = lane & 15;            // B/C/D column within tile
    const int khi = (lane >> 4) << 1;     // lanes 0-15: K+0/K+1 ; lanes 16-31: K+2/K+3

    const float* __restrict__ aptr = X + (size_t)(row0 + m) * DFEAT;

    v8f acc = {0.f, 0.f, 0.f, 0.f, 0.f, 0.f, 0.f, 0.f};

    for (int kb = 0; kb < DFEAT; kb += 4) {
        v2f a, b;
        a.x = aptr[kb + khi];
        a.y = aptr[kb + khi + 1];
        b.x = W[(size_t)(kb + khi)     * DFEAT + col0 + n];
        b.y = W[(size_t)(kb + khi + 1) * DFEAT + col0 + n];
        acc = __builtin_amdgcn_wmma_f32_16x16x4_f32(
            /*neg_a=*/false, a, /*neg_b=*/false, b,
            /*c_mod=*/(short)0, acc, /*reuse_a=*/false, /*reuse_b=*/false);
    }

    // C/D layout: VGPR i -> M = i (lanes 0-15) or i+8 (lanes 16-31), N = lane&15
    const int rhalf = (lane >> 4) * 8;
    float* __restrict__ out = H + (size_t)(row0 + rhalf) * DFEAT + col0 + n;
#pragma unroll
    for (int i = 0; i < 8; ++i) {
        out[(size_t)i * DFEAT] = acc[i];
    }
}

// ---------------- aggregation ----------------
// AGG[i,:] = bias + H[i,:] * dinv[i]^2  (self-loop term, norm = dinv[i]*dinv[i])
__global__ void agg_init_kernel(const float* __restrict__ H, const float* __restrict__ dinv,
                                const float* __restrict__ bias, float* __restrict__ AGG, int N) {
    int i = blockIdx.x * blockDim.x + threadIdx.x;
    if (i >= N * DFEAT) return;
    int row = i >> 7;
    int c   = i & (DFEAT - 1);
    float dv = dinv[row];
    AGG[i] = bias[c] + H[i] * dv * dv;
}

// One wave32 per edge: lane handles 4 features (float4 gather + 4 HW float atomics).
// Edge indices / norm are wave-uniform -> hoist to scalar regs with readfirstlane.
// Prefetch the next edge-group's source row to hide gather latency behind atomics.
#define PF_DIST 8
__global__ void edge_scatter_kernel(const float* __restrict__ H, const long long* __restrict__ ei,
                                    const float* __restrict__ dinv, float* __restrict__ AGG, int E) {
    int e    = blockIdx.x * (blockDim.x >> 5) + (threadIdx.x >> 5);
    int lane = threadIdx.x & 31;
    if (e >= E) return;

    int s = __builtin_amdgcn_readfirstlane((int)ei[e]);
    int d = __builtin_amdgcn_readfirstlane((int)ei[(size_t)E + e]);

    // prefetch the H row we will gather PF_DIST edge-groups from now
    int ep = e + PF_DIST;
    if (ep < E) {
        int sp = __builtin_amdgcn_readfirstlane((int)ei[ep]);
        __builtin_prefetch(H + (size_t)sp * DFEAT + lane * 4, 0, 3);
    }

    float w = dinv[s] * dinv[d];
    const float4 v = ((const float4*)(H + (size_t)s * DFEAT))[lane];
    float* __restrict__ out = AGG + (size_t)d * DFEAT + lane * 4;
    unsafeAtomicAdd(out + 0, v.x * w);
    unsafeAtomicAdd(out + 1, v.y * w);
    unsafeAtomicAdd(out + 2, v.z * w);
    unsafeAtomicAdd(out + 3, v.w * w);
}

// ---------------- batch norm ----------------
__global__ void bn_zero_kernel(float* __restrict__ sums, float* __restrict__ sumsq) {
    int i = threadIdx.x;
    if (i < DFEAT) { sums[i] = 0.f; sumsq[i] = 0.f; }
}

__global__ void bn_stats_kernel(const float* __restrict__ AGG, float* __restrict__ sums,
                                float* __restrict__ sumsq, int N) {
    __shared__ float s1[256];
    __shared__ float s2[256];
    const int c  = threadIdx.x & (DFEAT - 1);
    const int rl = threadIdx.x >> 7;           // 0 or 1: two rows in flight per block
    float a = 0.f, b = 0.f;
    for (int r = blockIdx.x * 2 + rl; r < N; r += gridDim.x * 2) {
        float v = AGG[(size_t)r * DFEAT + c];
        a += v;
        b += v * v;
    }
    s1[threadIdx.x] = a;
    s2[threadIdx.x] = b;
    __syncthreads();
    if (rl == 0) {
        unsafeAtomicAdd(&sums[c],  s1[c] + s1[c + DFEAT]);
        unsafeAtomicAdd(&sumsq[c], s2[c] + s2[c + DFEAT]);
    }
}

__global__ void bn_apply_kernel(const float* __restrict__ AGG, const float* __restrict__ sums,
                                const float* __restrict__ sumsq, const float* __restrict__ g,
                                const float* __restrict__ bt, float* __restrict__ OUT,
                                int N, int Nout) {
    int i = blockIdx.x * blockDim.x + threadIdx.x;
    if (i >= Nout * DFEAT) return;
    int c = i & (DFEAT - 1);
    float invn = 1.0f / (float)N;
    float mean = sums[c] * invn;
    float var  = sumsq[c] * invn - mean * mean;
    OUT[i] = (AGG[i] - mean) * rsqrtf(var + BN_EPS) * g[c] + bt[c];
}

// ---------------- host orchestration ----------------
extern "C" void kernel_launch(void* const* d_in, const int* in_sizes, int n_in,
                              void* d_out, int out_size, void* d_ws, size_t ws_size,
                              hipStream_t stream) {
    const float*     x  = (const float*)d_in[0];
    const long long* ei = (const long long*)d_in[1];   // (2,E) int64: [src | dst]
    const int N = in_sizes[0] / DFEAT;
    const int E = in_sizes[1] / 2;

    // workspace carve-up (floats)
    float* ws = (float*)d_ws;
    size_t off = 0;
    float* dinv = ws + off; off += (((size_t)N + 255) & ~(size_t)255);
    float* H    = ws + off; off += (size_t)N * DFEAT;
    float* AGG  = ws + off; off += (size_t)N * DFEAT;
    float* bufA = ws + off; off += (size_t)N * DFEAT;
    float* sums = ws + off; off += 128;
    float* sumsq = ws + off; off += 128;
    (void)ws_size; (void)n_in;

    // symmetric GCN normalization (computed once; graph is static across layers)
    deg_init_kernel<<<(N + 255) / 256, 256, 0, stream>>>(dinv, N);
    deg_accum_kernel<<<(E + 255) / 256, 256, 0, stream>>>(ei + E, dinv, E);
    dinv_kernel<<<(N + 255) / 256, 256, 0, stream>>>(dinv, N);

    const float* Xcur = x;
    const int NoutFinal = out_size / DFEAT;   // xdim rows

    for (int layer = 0; layer < 3; ++layer) {
        const float* Wl  = (const float*)d_in[3 + 4 * layer];
        const float* bl  = (const float*)d_in[4 + 4 * layer];
        const float* gl  = (const float*)d_in[5 + 4 * layer];
        const float* btl = (const float*)d_in[6 + 4 * layer];

        gemm_wmma_kernel<<<N / 16, 256, 0, stream>>>(Xcur, Wl, H);
        agg_init_kernel<<<(N * DFEAT + 255) / 256, 256, 0, stream>>>(H, dinv, bl, AGG, N);
        edge_scatter_kernel<<<(E + 7) / 8, 256, 0, stream>>>(H, ei, dinv, AGG, E);
        bn_zero_kernel<<<1, 128, 0, stream>>>(sums, sumsq);
        bn_stats_kernel<<<512, 256, 0, stream>>>(AGG, sums, sumsq, N);

        float* OUT = (layer == 2) ? (float*)d_out : bufA;
        int    no  = (layer == 2) ? NoutFinal : N;
        bn_apply_kernel<<<((size_t)no * DFEAT + 255) / 256, 256, 0, stream>>>(
            AGG, sums, sumsq, gl, btl, OUT, N, no);

        Xcur = bufA;
    }
}